// PGIN_71425306133016
// MI455X (gfx1250) — compile-verified
//
#include <hip/hip_runtime.h>

typedef float v2f __attribute__((ext_vector_type(2)));
typedef float v8f __attribute__((ext_vector_type(8)));

#define N_NODES 100000   // == 3125 * 32 : exact tiling, no tail guards needed
#define N_EDGES 1000000
#define IN_CH 128
#define HID 64
#define S_CH 32
#define N_LAYERS 3

// ---------------------------------------------------------------------------
// zero a float buffer (graph-capture safe; no hipMemset)
// ---------------------------------------------------------------------------
__global__ __launch_bounds__(256) void zero_f32(float* __restrict__ p, int n) {
    int stride = gridDim.x * blockDim.x;
    for (int i = blockIdx.x * blockDim.x + threadIdx.x; i < n; i += stride)
        p[i] = 0.0f;
}

// ---------------------------------------------------------------------------
// scatter-add message passing: agg[dst, :] += h[src, :] * ew
// one thread handles 4 channels of one edge (float4 gather + 4 f32 atomics;
// node arrays fit in the 192MB L2 so atomics resolve on-chip)
// ---------------------------------------------------------------------------
template <int K, int HS>
__global__ __launch_bounds__(256) void scatter_add(
    const float* __restrict__ h,
    const long long* __restrict__ src,
    const long long* __restrict__ dst,
    const float* __restrict__ ew,
    float* __restrict__ agg) {
    constexpr int K4 = K / 4;
    long long idx = (long long)blockIdx.x * blockDim.x + threadIdx.x;
    int e  = (int)(idx / K4);   // K4 is power of two -> shifts
    int c4 = (int)(idx % K4);
    int s = (int)src[e];
    int d = (int)dst[e];
    float w = ew[e];
    const float4 hv = *reinterpret_cast<const float4*>(h + (size_t)s * HS + c4 * 4);
    float* base = agg + (size_t)d * K + c4 * 4;
    atomicAdd(base + 0, hv.x * w);
    atomicAdd(base + 1, hv.y * w);
    atomicAdd(base + 2, hv.z * w);
    atomicAdd(base + 3, hv.w * w);
}

// ---------------------------------------------------------------------------
// fused GIN MLP for one layer, 32 nodes per block, 8 waves:
//   z = (1+eps)*h + agg          -> LDS
//   z @ W1 + b1                  -> WMMA f32 16x16x4 (K steps of 4)
//   BN(eval) + ReLU              -> in-register, per-lane channel
//   @ W2 + b2, ReLU              -> WMMA again
//   write hcat[:, layer*64 : +64]
// wave w: node sub-tile = (w>>2)*16, out-channel tile = (w&3)*16
// WMMA lane mapping: m/n = lane&15, koff = 2*(lane>>4), C/D row = v+8*(lane>>4)
// N_NODES % 32 == 0 -> no tail: all loads/stores unguarded, EXEC stays full.
// ---------------------------------------------------------------------------
template <int K>
__global__ __launch_bounds__(256) void gin_mlp(
    const float* __restrict__ h, int hstride,
    const float* __restrict__ agg,
    const float* __restrict__ eps_arr, int layer,
    const float* __restrict__ W1, const float* __restrict__ b1,
    const float* __restrict__ W2, const float* __restrict__ b2,
    const float* __restrict__ gamma, const float* __restrict__ beta,
    const float* __restrict__ mean, const float* __restrict__ var,
    float* __restrict__ hcat) {
    constexpr int KP = K + 4;     // pad LDS rows to avoid bank conflicts
    constexpr int HP = HID + 4;
    __shared__ float z_lds[32 * KP];
    __shared__ float h1_lds[32 * HP];

    const int tid  = threadIdx.x;
    const int lane = tid & 31;
    const int wave = tid >> 5;
    const int tileM = (wave >> 2) * 16;  // 0 or 16
    const int nOff  = (wave & 3) * 16;   // 0,16,32,48
    const int blockNode = blockIdx.x * 32;
    const float ep = 1.0f + eps_arr[layer];

    // stage z = (1+eps)*h + agg into LDS
    for (int i = tid; i < 32 * K; i += 256) {
        int r = i / K, c = i % K;  // K is power of two -> shifts
        int node = blockNode + r;
        z_lds[r * KP + c] = ep * h[(size_t)node * hstride + c] + agg[(size_t)node * K + c];
    }
    __syncthreads();

    const int m    = lane & 15;
    const int koff = (lane >> 4) * 2;

    // ---- GEMM1: [16 x K] @ [K x 16] via V_WMMA_F32_16X16X4_F32 ----
    v8f acc = {};
    for (int k = 0; k < K; k += 4) {
        v2f a, b;
        a.x = z_lds[(tileM + m) * KP + k + koff];
        a.y = z_lds[(tileM + m) * KP + k + koff + 1];
        b.x = W1[(k + koff) * HID + nOff + m];
        b.y = W1[(k + koff + 1) * HID + nOff + m];
        acc = __builtin_amdgcn_wmma_f32_16x16x4_f32(
            false, a, false, b, (short)0, acc, false, false);
    }

    // ---- bias + BatchNorm(eval) + ReLU, stage to LDS ----
    {
        int ch = nOff + m;
        float sc = gamma[ch] * rsqrtf(var[ch] + 1e-5f);
        float sh = beta[ch] - mean[ch] * sc;
        float bb = b1[ch];
#pragma unroll
        for (int v = 0; v < 8; ++v) {
            int row = tileM + v + 8 * (lane >> 4);
            float t = (acc[v] + bb) * sc + sh;
            t = t > 0.0f ? t : 0.0f;
            h1_lds[row * HP + ch] = t;
        }
    }
    __syncthreads();

    // ---- GEMM2: [16 x 64] @ [64 x 16] ----
    v8f acc2 = {};
    for (int k = 0; k < HID; k += 4) {
        v2f a, b;
        a.x = h1_lds[(tileM + m) * HP + k + koff];
        a.y = h1_lds[(tileM + m) * HP + k + koff + 1];
        b.x = W2[(k + koff) * HID + nOff + m];
        b.y = W2[(k + koff + 1) * HID + nOff + m];
        acc2 = __builtin_amdgcn_wmma_f32_16x16x4_f32(
            false, a, false, b, (short)0, acc2, false, false);
    }

    // ---- bias + ReLU, write into hcat slice (next layer's h), no guards ----
    {
        int ch = nOff + m;
        float bb = b2[ch];
#pragma unroll
        for (int v = 0; v < 8; ++v) {
            int node = blockNode + tileM + v + 8 * (lane >> 4);
            float t = acc2[v] + bb;
            t = t > 0.0f ? t : 0.0f;
            hcat[(size_t)node * (HID * N_LAYERS) + layer * HID + ch] = t;
        }
    }
}

// ---------------------------------------------------------------------------
// final projection: out = hcat[N,192] @ lin2_W[192,32] + lin2_b
// 32 nodes per block (4 waves = 2 node tiles x 2 channel tiles), 3125 blocks
// exact -> no guards. A rows read straight from global (L2-resident).
// ---------------------------------------------------------------------------
__global__ __launch_bounds__(128) void final_proj(
    const float* __restrict__ hcat,
    const float* __restrict__ W, const float* __restrict__ bvec,
    float* __restrict__ out) {
    const int tid  = threadIdx.x;
    const int lane = tid & 31;
    const int wave = tid >> 5;
    const int tileM = (wave >> 1) * 16;  // 0 or 16
    const int nOff  = (wave & 1) * 16;   // 0 or 16
    const int blockNode = blockIdx.x * 32;
    const int m    = lane & 15;
    const int koff = (lane >> 4) * 2;
    constexpr int K = HID * N_LAYERS;  // 192

    const float* arow = hcat + (size_t)(blockNode + tileM + m) * K;

    v8f acc = {};
    for (int k = 0; k < K; k += 4) {
        v2f a, b;
        a.x = arow[k + koff];
        a.y = arow[k + koff + 1];
        b.x = W[(k + koff) * S_CH + nOff + m];
        b.y = W[(k + koff + 1) * S_CH + nOff + m];
        acc = __builtin_amdgcn_wmma_f32_16x16x4_f32(
            false, a, false, b, (short)0, acc, false, false);
    }
    float bias = bvec[nOff + m];
#pragma unroll
    for (int v = 0; v < 8; ++v) {
        int node = blockNode + tileM + v + 8 * (lane >> 4);
        out[(size_t)node * S_CH + nOff + m] = acc[v] + bias;
    }
}

// ---------------------------------------------------------------------------
extern "C" void kernel_launch(void* const* d_in, const int* in_sizes, int n_in,
                              void* d_out, int out_size, void* d_ws, size_t ws_size,
                              hipStream_t stream) {
    const float*     x      = (const float*)d_in[0];
    const long long* ei     = (const long long*)d_in[1];  // int64 [2,E]
    const float*     ew     = (const float*)d_in[2];
    const float*     eps    = (const float*)d_in[3];
    const float*     W1_0   = (const float*)d_in[4];
    const float*     b1_0   = (const float*)d_in[5];
    const float*     W1_r   = (const float*)d_in[6];
    const float*     b1_r   = (const float*)d_in[7];
    const float*     W2     = (const float*)d_in[8];
    const float*     b2     = (const float*)d_in[9];
    const float*     gamma  = (const float*)d_in[10];
    const float*     beta   = (const float*)d_in[11];
    const float*     mean   = (const float*)d_in[12];
    const float*     var    = (const float*)d_in[13];
    const float*     lin2_W = (const float*)d_in[14];
    const float*     lin2_b = (const float*)d_in[15];
    float* out = (float*)d_out;

    // workspace layout: agg [N,128] f32 | hcat [N,192] f32  (~128 MB total)
    float* agg  = (float*)d_ws;
    float* hcat = agg + (size_t)N_NODES * IN_CH;

    const long long* src = ei;
    const long long* dst = ei + N_EDGES;

    for (int layer = 0; layer < N_LAYERS; ++layer) {
        if (layer == 0) {
            zero_f32<<<2048, 256, 0, stream>>>(agg, N_NODES * IN_CH);
            long long tot = (long long)N_EDGES * (IN_CH / 4);  // 32M, exact /256
            scatter_add<IN_CH, IN_CH>
                <<<(unsigned)(tot / 256), 256, 0, stream>>>(x, src, dst, ew, agg);
            gin_mlp<IN_CH><<<N_NODES / 32, 256, 0, stream>>>(
                x, IN_CH, agg, eps, 0,
                W1_0, b1_0, W2, b2, gamma, beta, mean, var, hcat);
        } else {
            const float* h = hcat + (size_t)(layer - 1) * HID;  // row stride 192
            zero_f32<<<2048, 256, 0, stream>>>(agg, N_NODES * HID);
            long long tot = (long long)N_EDGES * (HID / 4);     // 16M, exact /256
            scatter_add<HID, HID * N_LAYERS>
                <<<(unsigned)(tot / 256), 256, 0, stream>>>(h, src, dst, ew, agg);
            gin_mlp<HID><<<N_NODES / 32, 256, 0, stream>>>(
                h, HID * N_LAYERS, agg, eps, layer,
                W1_r + (size_t)(layer - 1) * HID * HID, b1_r + (size_t)(layer - 1) * HID,
                W2 + (size_t)layer * HID * HID, b2 + (size_t)layer * HID,
                gamma + (size_t)layer * HID, beta + (size_t)layer * HID,
                mean + (size_t)layer * HID, var + (size_t)layer * HID, hcat);
        }
    }
    final_proj<<<N_NODES / 32, 128, 0, stream>>>(hcat, lin2_W, lin2_b, out);
}